// ProposalLayer_86517821214886
// MI455X (gfx1250) — compile-verified
//
#include <hip/hip_runtime.h>
#include <math.h>

// ---------------------------------------------------------------------------
// ProposalLayer for MI455X (gfx1250, wave32).
// No matmul in this op -> WMMA not applicable; the CDNA5-specific path used is
// the async-to-LDS datapath (global_load_async_to_lds_b128 / s_wait_asynccnt)
// plus wave32-sized (32-thread) NMS tiles.
// Workspace requirement: ~7.1 MB.
// ---------------------------------------------------------------------------

#define NPROP    589824      // 256*256*9
#define HW       65536       // 256*256
#define WDIM     256
#define A_NUM    9
#define K_PRE    6000
#define K_POST   300
#define CANDMAX  8192
#define CB       188         // ceil(6000/32) suppression-mask words per row
#define NBOX_PAD 6016        // 6000 padded to a multiple of 32

__device__ __forceinline__ float clampf(float v, float lo, float hi) {
  return fminf(fmaxf(v, lo), hi);
}

// Decode proposal p -> clipped box (matches reference math exactly).
__device__ __forceinline__ void decode_box(const float* __restrict__ deltas,
                                           const float* __restrict__ anchors,
                                           const float* __restrict__ img_info,
                                           int p, float& x1, float& y1,
                                           float& x2, float& y2) {
  int a    = p % A_NUM;
  int cell = p / A_NUM;
  int w    = cell & (WDIM - 1);
  int h    = cell >> 8;
  int base = (a * 4) * HW + h * WDIM + w;
  float dx = deltas[base];
  float dy = deltas[base + HW];
  float dw = clampf(deltas[base + 2 * HW], -10.0f, 10.0f);
  float dh = clampf(deltas[base + 3 * HW], -10.0f, 10.0f);
  float sx = (float)w * 16.0f, sy = (float)h * 16.0f;
  float ax1 = anchors[a * 4 + 0] + sx;
  float ay1 = anchors[a * 4 + 1] + sy;
  float ax2 = anchors[a * 4 + 2] + sx;
  float ay2 = anchors[a * 4 + 3] + sy;
  float aw  = ax2 - ax1 + 1.0f;
  float ah  = ay2 - ay1 + 1.0f;
  float acx = ax1 + 0.5f * aw;
  float acy = ay1 + 0.5f * ah;
  float pcx = dx * aw + acx;
  float pcy = dy * ah + acy;
  float pw  = expf(dw) * aw;
  float ph  = expf(dh) * ah;
  float Wm1 = img_info[1] - 1.0f;
  float Hm1 = img_info[0] - 1.0f;
  x1 = clampf(pcx - 0.5f * pw, 0.0f, Wm1);
  y1 = clampf(pcy - 0.5f * ph, 0.0f, Hm1);
  x2 = clampf(pcx + 0.5f * pw - 1.0f, 0.0f, Wm1);
  y2 = clampf(pcy + 0.5f * ph - 1.0f, 0.0f, Hm1);
}

// ---------------------------------------------------------------------------
__global__ void init_kernel(unsigned* hist1, unsigned* hist2, unsigned* sel,
                            float4* topBoxes) {
  int t = blockIdx.x * blockDim.x + threadIdx.x;
  if (t < 2048) { hist1[t] = 0u; hist2[t] = 0u; }
  if (t < 8) sel[t] = 0u;
  if (t < (NBOX_PAD - K_PRE)) topBoxes[K_PRE + t] = make_float4(0.f, 0.f, 0.f, 0.f);
}

// Stage 1: decode, min-size filter, map masked score -> monotonic u32 key.
__global__ void decode_key_kernel(const float* __restrict__ probs,
                                  const float* __restrict__ deltas,
                                  const float* __restrict__ img_info,
                                  const float* __restrict__ anchors,
                                  unsigned* __restrict__ keys) {
  int p = blockIdx.x * blockDim.x + threadIdx.x;
  if (p >= NPROP) return;
  float x1, y1, x2, y2;
  decode_box(deltas, anchors, img_info, p, x1, y1, x2, y2);
  int a    = p % A_NUM;
  int cell = p / A_NUM;
  int w    = cell & (WDIM - 1);
  int h    = cell >> 8;
  float scr = probs[(A_NUM + a) * HW + h * WDIM + w];   // fg slice
  float ms  = 16.0f * img_info[2];
  bool valid = (x2 - x1 + 1.0f >= ms) && (y2 - y1 + 1.0f >= ms);
  float s = valid ? scr : -INFINITY;
  unsigned bits = __float_as_uint(s);
  unsigned key  = bits ^ ((bits & 0x80000000u) ? 0xFFFFFFFFu : 0x80000000u);
  keys[p] = key;   // all real scores >= 0 -> key >= 0x80000000; -inf -> 0x007FFFFF
}

// Stage 2a: 2048-bucket histogram of key[31:21].
__global__ void hist1_kernel(const unsigned* __restrict__ keys, unsigned* hist) {
  __shared__ unsigned h[2048];
  for (int b = threadIdx.x; b < 2048; b += blockDim.x) h[b] = 0u;
  __syncthreads();
  for (int p = blockIdx.x * blockDim.x + threadIdx.x; p < NPROP;
       p += gridDim.x * blockDim.x)
    atomicAdd(&h[keys[p] >> 21], 1u);
  __syncthreads();
  for (int b = threadIdx.x; b < 2048; b += blockDim.x)
    if (h[b]) atomicAdd(&hist[b], h[b]);
}

__global__ void select1_kernel(const unsigned* __restrict__ hist, unsigned* sel) {
  if (threadIdx.x != 0) return;
  unsigned acc = 0; unsigned b1 = 0; unsigned above = 0; bool found = false;
  for (int b = 2047; b >= 0; --b) {
    unsigned c = hist[b];
    if (!found && acc + c >= K_PRE) { b1 = (unsigned)b; above = acc; found = true; break; }
    acc += c;
  }
  if (!found) { b1 = 0u; above = acc; }
  sel[0] = b1; sel[1] = above;
}

// Stage 2b: refine within bucket B1 on key[20:10].
__global__ void hist2_kernel(const unsigned* __restrict__ keys,
                             const unsigned* __restrict__ sel, unsigned* hist) {
  __shared__ unsigned h[2048];
  unsigned B1 = sel[0];
  for (int b = threadIdx.x; b < 2048; b += blockDim.x) h[b] = 0u;
  __syncthreads();
  for (int p = blockIdx.x * blockDim.x + threadIdx.x; p < NPROP;
       p += gridDim.x * blockDim.x) {
    unsigned key = keys[p];
    if ((key >> 21) == B1) atomicAdd(&h[(key >> 10) & 0x7FFu], 1u);
  }
  __syncthreads();
  for (int b = threadIdx.x; b < 2048; b += blockDim.x)
    if (h[b]) atomicAdd(&hist[b], h[b]);
}

__global__ void select2_kernel(const unsigned* __restrict__ hist, unsigned* sel) {
  if (threadIdx.x != 0) return;
  unsigned acc = sel[1]; unsigned b2 = 0; bool found = false;
  for (int b = 2047; b >= 0; --b) {
    unsigned c = hist[b];
    if (acc + c >= K_PRE) { b2 = (unsigned)b; found = true; break; }
    acc += c;
  }
  if (!found) b2 = 0u;
  sel[2] = (sel[0] << 11) | b2;   // 22-bit threshold on key>>10
  sel[3] = 0u;                    // candidate counter (reset every call)
}

// Stage 2c: compact all keys at/above the 22-bit threshold.
__global__ void collect_kernel(const unsigned* __restrict__ keys,
                               unsigned* __restrict__ sel,
                               unsigned long long* __restrict__ cand) {
  unsigned T = sel[2];
  for (int p = blockIdx.x * blockDim.x + threadIdx.x; p < NPROP;
       p += gridDim.x * blockDim.x) {
    unsigned key = keys[p];
    if ((key >> 10) >= T) {
      unsigned pos = atomicAdd(&sel[3], 1u);
      if (pos < CANDMAX)
        cand[pos] = ((unsigned long long)key << 32) |
                    (unsigned long long)(0xFFFFFFFFu - (unsigned)p);
    }
  }
}

// Stage 2d: bitonic-sort <=8192 candidates descending in LDS (64 KB of the
// 320 KB/WGP LDS); emit sorted top-6000 (key + original index).
__global__ void sort_topk_kernel(const unsigned long long* __restrict__ cand,
                                 const unsigned* __restrict__ sel,
                                 unsigned* __restrict__ topKey,
                                 unsigned* __restrict__ order) {
  __shared__ unsigned long long s[CANDMAX];
  int t = threadIdx.x;
  unsigned count = sel[3]; if (count > CANDMAX) count = CANDMAX;
  for (int i = t; i < CANDMAX; i += blockDim.x)
    s[i] = (i < (int)count) ? cand[i] : 0ull;
  __syncthreads();
  for (int k = 2; k <= CANDMAX; k <<= 1) {
    for (int j = k >> 1; j > 0; j >>= 1) {
      for (int i = t; i < CANDMAX; i += blockDim.x) {
        int ixj = i ^ j;
        if (ixj > i) {
          unsigned long long a = s[i], b = s[ixj];
          bool descBlock = ((i & k) == 0);
          if (descBlock ? (a < b) : (a > b)) { s[i] = b; s[ixj] = a; }
        }
      }
      __syncthreads();
    }
  }
  for (int r = t; r < K_PRE; r += blockDim.x) {
    unsigned long long v = s[r];
    topKey[r] = (unsigned)(v >> 32);
    order[r]  = 0xFFFFFFFFu - (unsigned)(v & 0xFFFFFFFFull);
  }
}

// Stage 3: re-decode the 6000 selected boxes (cheaper than storing 589K boxes).
__global__ void gather_boxes_kernel(const unsigned* __restrict__ topKey,
                                    const unsigned* __restrict__ order,
                                    const float* __restrict__ deltas,
                                    const float* __restrict__ img_info,
                                    const float* __restrict__ anchors,
                                    float4* __restrict__ topBoxes) {
  int r = blockIdx.x * blockDim.x + threadIdx.x;
  if (r >= K_PRE) return;
  float4 b = make_float4(0.f, 0.f, 0.f, 0.f);
  if (topKey[r] >= 0x80000000u) {   // real (non -inf, non padding) entry
    float x1, y1, x2, y2;
    decode_box(deltas, anchors, img_info, (int)order[r], x1, y1, x2, y2);
    b = make_float4(x1, y1, x2, y2);
  }
  topBoxes[r] = b;
}

// Stage 4: pairwise suppression bitmask. 32 threads = one wave32 per 32x32
// tile; the 32 column boxes (512 B) are copied into LDS through the CDNA5
// async-to-LDS path (tracked by ASYNCcnt, s_wait_asynccnt).
__global__ void nms_mask_kernel(const float4* __restrict__ topBoxes,
                                unsigned* __restrict__ mask) {
  __shared__ float sbox[32 * 4];    // sole LDS object in this kernel -> offset 0
  int jb = blockIdx.x, ib = blockIdx.y;
  int t  = threadIdx.x;
  {
    const float4* gsrc = topBoxes + (jb * 32 + t);  // padded to 6016 entries
    unsigned ldsoff = (unsigned)(t * 16);           // sbox base is LDS offset 0
    asm volatile("global_load_async_to_lds_b128 %0, %1, off"
                 :: "v"(ldsoff), "v"(gsrc) : "memory");
    asm volatile("s_wait_asynccnt 0" ::: "memory");
  }
  __syncthreads();
  int i = ib * 32 + t;
  unsigned bits = 0u;
  if (i < K_PRE && jb >= ib) {      // jb < ib tiles have only j < i -> all zero
    float4 bi = topBoxes[i];
    float ai  = (bi.z - bi.x + 1.0f) * (bi.w - bi.y + 1.0f);
    volatile const float* sb = sbox;
    for (int jj = 0; jj < 32; ++jj) {
      int j = jb * 32 + jj;
      float bx1 = sb[jj * 4 + 0], by1 = sb[jj * 4 + 1];
      float bx2 = sb[jj * 4 + 2], by2 = sb[jj * 4 + 3];
      float aj  = (bx2 - bx1 + 1.0f) * (by2 - by1 + 1.0f);
      float xx1 = fmaxf(bi.x, bx1), yy1 = fmaxf(bi.y, by1);
      float xx2 = fminf(bi.z, bx2), yy2 = fminf(bi.w, by2);
      float iw = fmaxf(xx2 - xx1 + 1.0f, 0.0f);
      float ih = fmaxf(yy2 - yy1 + 1.0f, 0.0f);
      float inter = iw * ih;
      bool sup = inter > 0.7f * (ai + aj - inter);   // iou > NMS_THRESH
      if (sup && j > i && j < K_PRE) bits |= (1u << jj);
    }
  }
  if (i < K_PRE) mask[i * CB + jb] = bits;
}

// Stage 5: sequential greedy scan (matches reference fori_loop order) + output.
__global__ void nms_reduce_kernel(const unsigned* __restrict__ topKey,
                                  const float4* __restrict__ topBoxes,
                                  const unsigned* __restrict__ mask,
                                  float* __restrict__ out) {
  __shared__ unsigned removed[CB];
  int t = threadIdx.x;                      // 192 threads (6 waves)
  if (t < CB) {
    unsigned w = 0u;
    for (int b = 0; b < 32; ++b) {
      int s = t * 32 + b;
      bool dead = (s >= K_PRE) || (topKey[s] < 0x80000000u);  // -inf / padding
      if (dead) w |= (1u << b);
    }
    removed[t] = w;
  }
  __syncthreads();
  for (int i = 0; i < K_PRE; ++i) {
    bool alive = ((removed[i >> 5] >> (i & 31)) & 1u) == 0u;
    __syncthreads();
    if (alive && t < CB) removed[t] |= mask[i * CB + t];
    __syncthreads();
  }
  if (t == 0) {
    int cnt = 0;
    for (int s = 0; s < K_PRE && cnt < K_POST; ++s) {
      if (!((removed[s >> 5] >> (s & 31)) & 1u)) {
        float4 b = topBoxes[s];
        out[cnt * 5 + 0] = 0.0f;
        out[cnt * 5 + 1] = b.x;
        out[cnt * 5 + 2] = b.y;
        out[cnt * 5 + 3] = b.z;
        out[cnt * 5 + 4] = b.w;
        ++cnt;
      }
    }
    for (; cnt < K_POST; ++cnt) {
      out[cnt * 5 + 0] = 0.0f; out[cnt * 5 + 1] = 0.0f; out[cnt * 5 + 2] = 0.0f;
      out[cnt * 5 + 3] = 0.0f; out[cnt * 5 + 4] = 0.0f;
    }
  }
}

// ---------------------------------------------------------------------------
extern "C" void kernel_launch(void* const* d_in, const int* in_sizes, int n_in,
                              void* d_out, int out_size, void* d_ws, size_t ws_size,
                              hipStream_t stream) {
  (void)in_sizes; (void)n_in; (void)out_size; (void)ws_size;
  const float* probs    = (const float*)d_in[0];  // (1,18,256,256)
  const float* deltas   = (const float*)d_in[1];  // (1,36,256,256)
  const float* img_info = (const float*)d_in[2];  // (3,)
  const float* anchors  = (const float*)d_in[3];  // (9,4)
  float* out = (float*)d_out;                     // (300,5) fp32

  // Workspace carve-up (16B aligned), total ~7.1 MB.
  char* W = (char*)d_ws;
  unsigned*           keys     = (unsigned*)(W + 0);            // 589824 u32
  unsigned long long* cand     = (unsigned long long*)(W + 2359296); // 8192 u64
  unsigned*           topKey   = (unsigned*)(W + 2424832);      // 6000 u32
  unsigned*           order    = (unsigned*)(W + 2448832);      // 6000 u32
  float4*             topBoxes = (float4*)(W + 2472832);        // 6016 float4
  unsigned*           mask     = (unsigned*)(W + 2569088);      // 6000*188 u32
  unsigned*           hist1    = (unsigned*)(W + 7081088);      // 2048 u32
  unsigned*           hist2    = (unsigned*)(W + 7089280);      // 2048 u32
  unsigned*           sel      = (unsigned*)(W + 7097472);      // 8 u32

  init_kernel<<<2, 1024, 0, stream>>>(hist1, hist2, sel, topBoxes);
  decode_key_kernel<<<(NPROP + 255) / 256, 256, 0, stream>>>(
      probs, deltas, img_info, anchors, keys);
  hist1_kernel<<<512, 256, 0, stream>>>(keys, hist1);
  select1_kernel<<<1, 32, 0, stream>>>(hist1, sel);
  hist2_kernel<<<512, 256, 0, stream>>>(keys, sel, hist2);
  select2_kernel<<<1, 32, 0, stream>>>(hist2, sel);
  collect_kernel<<<512, 256, 0, stream>>>(keys, sel, cand);
  sort_topk_kernel<<<1, 1024, 0, stream>>>(cand, sel, topKey, order);
  gather_boxes_kernel<<<(K_PRE + 255) / 256, 256, 0, stream>>>(
      topKey, order, deltas, img_info, anchors, topBoxes);
  nms_mask_kernel<<<dim3(CB, CB), 32, 0, stream>>>(topBoxes, mask);
  nms_reduce_kernel<<<1, 192, 0, stream>>>(topKey, topBoxes, mask, out);
}